// Conv1D_50113678409896
// MI455X (gfx1250) — compile-verified
//
#include <hip/hip_runtime.h>

// Problem constants (from reference setup_inputs): B=4, S=4096, W=2560, TW=4.
constexpr int B  = 4;
constexpr int S  = 4096;
constexpr int W  = 2560;
constexpr int TW = 4;

constexpr int BDIM = 128;          // 4 wave32s per block
constexpr int T    = 16;           // time steps per thread (x read amp = 19/16)
constexpr int NM   = T + TW - 1;   // masks needed per block = 19
constexpr int W4   = W / 4;        // 640 float4 channel groups per row

typedef float v4f __attribute__((ext_vector_type(4)));  // native vector for NT builtin

__device__ __forceinline__ float4 ld4(const float* p) {
  return *reinterpret_cast<const float4*>(p);
}

// Non-temporal 128-bit store: out is write-once/never-read, so keep it out of
// L2 (192 MB) and preserve capacity for the x halo rows that ARE reused.
__device__ __forceinline__ void st4_nt(float* p, float4 v) {
  v4f t = {v.x, v.y, v.z, v.w};
  __builtin_nontemporal_store(t, reinterpret_cast<v4f*>(p));
}

__global__ __launch_bounds__(BDIM) void conv1d_main_kernel(
    const float* __restrict__ x,    // (B,S,W)
    const int*   __restrict__ seg,  // (B,S)
    const float* __restrict__ w,    // (TW,W)
    const float* __restrict__ bias, // (W,)
    float*       __restrict__ out)  // (B,S,W)
{
  __shared__ int   sseg[NM];
  __shared__ float smask[NM];

  const int tid = threadIdx.x;
  const int c4  = blockIdx.x * BDIM + tid;   // channel-group 0..W4-1
  const int b   = blockIdx.z;
  const int s0  = blockIdx.y * T;

  // --- CDNA5 async global->LDS staging of the per-row document masks.
  // All 128 threads of this block need the same 19 segment_pos values;
  // lanes 0..18 pull them into LDS via the ASYNCcnt-tracked async copy path.
  if (tid < NM) {
    const int row  = s0 - (TW - 1) + tid;
    const int rowc = row < 0 ? 0 : row;      // clamp (masked to 0 below)
    unsigned long long gaddr =
        (unsigned long long)(const void*)(seg + (size_t)b * S + rowc);
    unsigned lds_addr = (unsigned)(unsigned long long)(const void*)&sseg[tid];
    asm volatile("global_load_async_to_lds_b32 %0, %1, off"
                 :
                 : "v"(lds_addr), "v"(gaddr)
                 : "memory");
    asm volatile("s_wait_asynccnt 0x0" ::: "memory");
    smask[tid] = (row >= 0 && sseg[tid] != 0) ? 1.0f : 0.0f;
  }
  __syncthreads();

  const int c = c4 * 4;
  const float* xb = x   + (size_t)b * S * W + c;
  float*       ob = out + (size_t)b * S * W + c;

  // Per-thread taps + bias (50 KB total; L2-resident, broadcast across blocks).
  const float4 w0 = ld4(w + 0 * W + c);
  const float4 w1 = ld4(w + 1 * W + c);
  const float4 w2 = ld4(w + 2 * W + c);
  const float4 w3 = ld4(w + 3 * W + c);
  const float4 bv = ld4(bias + c);

  auto ldrow = [&](int row, float m) -> float4 {
    const int rowc = row < 0 ? 0 : row;
    float4 v = ld4(xb + (size_t)rowc * W);   // global_load_b128 (RT: halo reuse)
    return make_float4(v.x * m, v.y * m, v.z * m, v.w * m);
  };

  // Sliding window of the last 4 masked rows (register-resident).
  float4 r0 = ldrow(s0 - 3, smask[0]);
  float4 r1 = ldrow(s0 - 2, smask[1]);
  float4 r2 = ldrow(s0 - 1, smask[2]);

#pragma unroll
  for (int t = 0; t < T; ++t) {
    float4 r3 = ldrow(s0 + t, smask[t + 3]);
    float4 o;
    o.x = fmaf(r0.x, w0.x, fmaf(r1.x, w1.x, fmaf(r2.x, w2.x, fmaf(r3.x, w3.x, bv.x))));
    o.y = fmaf(r0.y, w0.y, fmaf(r1.y, w1.y, fmaf(r2.y, w2.y, fmaf(r3.y, w3.y, bv.y))));
    o.z = fmaf(r0.z, w0.z, fmaf(r1.z, w1.z, fmaf(r2.z, w2.z, fmaf(r3.z, w3.z, bv.z))));
    o.w = fmaf(r0.w, w0.w, fmaf(r1.w, w1.w, fmaf(r2.w, w2.w, fmaf(r3.w, w3.w, bv.w))));
    st4_nt(ob + (size_t)(s0 + t) * W, o);    // global_store_b128 th:NT
    r0 = r1; r1 = r2; r2 = r3;
  }
}

// new_cache = (x * doc_mask)[:, S-(TW-1):, :]  -> (B, TW-1, W)
__global__ __launch_bounds__(256) void conv1d_cache_kernel(
    const float* __restrict__ x,
    const int*   __restrict__ seg,
    float*       __restrict__ cache)  // (B, TW-1, W)
{
  const int idx   = blockIdx.x * blockDim.x + threadIdx.x;
  const int total = B * (TW - 1) * W4;
  if (idx >= total) return;

  const int c4 = idx % W4;
  const int r  = (idx / W4) % (TW - 1);
  const int b  = idx / (W4 * (TW - 1));
  const int row = S - (TW - 1) + r;

  const float m = (seg[(size_t)b * S + row] != 0) ? 1.0f : 0.0f;
  float4 v = ld4(x + ((size_t)b * S + row) * W + c4 * 4);
  v.x *= m; v.y *= m; v.z *= m; v.w *= m;
  st4_nt(cache + ((size_t)b * (TW - 1) + r) * W + c4 * 4, v);
}

extern "C" void kernel_launch(void* const* d_in, const int* in_sizes, int n_in,
                              void* d_out, int out_size, void* d_ws, size_t ws_size,
                              hipStream_t stream) {
  const float* x    = (const float*)d_in[0];
  const int*   seg  = (const int*)d_in[1];
  const float* w    = (const float*)d_in[2];
  const float* bias = (const float*)d_in[3];
  float* out = (float*)d_out;

  // Main conv output: (B,S,W) floats, then cache (B,TW-1,W) appended flat.
  dim3 grid(W4 / BDIM, S / T, B);   // (5, 256, 4)
  conv1d_main_kernel<<<grid, BDIM, 0, stream>>>(x, seg, w, bias, out);

  const int cache_elems = B * (TW - 1) * W4;  // 7680 float4 threads
  conv1d_cache_kernel<<<(cache_elems + 255) / 256, 256, 0, stream>>>(
      x, seg, out + (size_t)B * S * W);
}